// SparseGPModel_24953759990009
// MI455X (gfx1250) — compile-verified
//
#include <hip/hip_runtime.h>
#include <math.h>

// ---------------- CDNA5 WMMA types ----------------
typedef __attribute__((ext_vector_type(2))) float v2f;
typedef __attribute__((ext_vector_type(8))) float v8f;

#define JITTER   1e-6f
#define TWO_PI_F 6.283185307179586f

// ---------------- RBF kernels ----------------
// Kuu[i,j] = os * exp(-0.5*||(Z_i - Z_j)/ls||^2) + jitter*I
__global__ void rbf_kuu_kernel(const float* __restrict__ Z, float* __restrict__ Kuu,
                               const float* __restrict__ ls_p, const float* __restrict__ os_p,
                               int m, int d) {
  int idx = blockIdx.x * blockDim.x + threadIdx.x;
  if (idx >= m * m) return;
  int i = idx / m, j = idx % m;
  float inv_ls = 1.0f / ls_p[0];
  float sq = 0.0f;
#pragma unroll 8
  for (int t = 0; t < d; ++t) {
    float diff = (Z[i * d + t] - Z[j * d + t]) * inv_ls;
    sq += diff * diff;
  }
  sq = fmaxf(sq, 0.0f);
  float v = os_p[0] * expf(-0.5f * sq);
  if (i == j) v += JITTER;
  Kuu[(size_t)i * m + j] = v;
}

// Kuf[u,i] = os * exp(-0.5*||(Z_u - X_i)/ls||^2)   (m x n)
__global__ void rbf_kuf_kernel(const float* __restrict__ Z, const float* __restrict__ X,
                               float* __restrict__ Kuf,
                               const float* __restrict__ ls_p, const float* __restrict__ os_p,
                               int m, int n, int d) {
  long long idx = (long long)blockIdx.x * blockDim.x + threadIdx.x;
  if (idx >= (long long)m * n) return;
  int u = (int)(idx / n), i = (int)(idx % n);
  float inv_ls = 1.0f / ls_p[0];
  float sq = 0.0f;
#pragma unroll 8
  for (int t = 0; t < d; ++t) {
    float diff = (Z[u * d + t] - X[i * d + t]) * inv_ls;
    sq += diff * diff;
  }
  sq = fmaxf(sq, 0.0f);
  Kuf[(size_t)u * n + i] = os_p[0] * expf(-0.5f * sq);
}

// ---------------- In-place Cholesky (single workgroup, right-looking) ----------------
__global__ void cholesky_kernel(float* __restrict__ A, int m) {
  __shared__ float s_dk;
  for (int k = 0; k < m; ++k) {
    if (threadIdx.x == 0) s_dk = sqrtf(A[(size_t)k * m + k]);
    __syncthreads();
    float dk = s_dk;
    for (int i = k + 1 + threadIdx.x; i < m; i += blockDim.x)
      A[(size_t)i * m + k] /= dk;
    if (threadIdx.x == 0) A[(size_t)k * m + k] = dk;
    __syncthreads();
    // trailing update on lower triangle: A[i,j] -= A[i,k]*A[j,k], k < j <= i
    for (int i = k + 1; i < m; ++i) {
      float aik = A[(size_t)i * m + k];
      for (int j = k + 1 + threadIdx.x; j <= i; j += blockDim.x)
        A[(size_t)i * m + j] -= aik * A[(size_t)j * m + k];
    }
    __syncthreads();
  }
}

// ---------------- Lower-triangular inverse: one wave (32 lanes) per column ----------------
__global__ void trinv_kernel(const float* __restrict__ L, float* __restrict__ Linv, int m) {
  extern __shared__ float x[];   // m floats
  int j = blockIdx.x;
  int lane = threadIdx.x;        // blockDim.x == 32 (one wave32)
  if (lane == 0) x[j] = 1.0f / L[(size_t)j * m + j];
  __syncthreads();
  for (int i = j + 1; i < m; ++i) {
    float s = 0.0f;
    for (int k = j + lane; k < i; k += 32)
      s += L[(size_t)i * m + k] * x[k];
#pragma unroll
    for (int off = 16; off > 0; off >>= 1)
      s += __shfl_down(s, off, 32);
    if (lane == 0) x[i] = -s / L[(size_t)i * m + i];
    __syncthreads();
  }
  for (int i = lane; i < m; i += 32)
    Linv[(size_t)i * m + j] = (i < j) ? 0.0f : x[i];
}

// ---------------- f32 WMMA GEMM: C(MxN) = A(MxK) @ B(KxN), all row-major ----------------
// One wave per 64x16 C block (4 stacked 16x16 WMMA tiles). The strided B fragment
// (2 scalar loads) is reused by 4 WMMAs per k-step; A fragments are contiguous float2.
// Requires M % 64 == 0, N % 16 == 0, K % 4 == 0.
__global__ __launch_bounds__(256)
void wmma_gemm_f32_kernel(const float* __restrict__ A, const float* __restrict__ B,
                          float* __restrict__ C, int M, int N, int K) {
  int wave = threadIdx.x >> 5;
  int lane = threadIdx.x & 31;
  int tiles_n = N >> 4;
  int blocks_m = M >> 6;                    // 64-row blocks
  int total_tasks = blocks_m * tiles_n;
  int task = blockIdx.x * 8 + wave;
  if (task >= total_tasks) return;          // uniform per wave: EXEC stays all-1s for WMMA
  int bm = task / tiles_n;
  int tn = task - bm * tiles_n;
  int row0 = bm << 6, col0 = tn << 4;
  int half = lane >> 4;                     // 0: K pair {0,1}, 1: K pair {2,3}
  int l16  = lane & 15;

  v8f acc0 = {}, acc1 = {}, acc2 = {}, acc3 = {};
  const float* Arow = A + (size_t)(row0 + l16) * K + 2 * half;
  const size_t rstep = (size_t)16 * K;      // 16 rows down
  for (int k = 0; k < K; k += 4) {
    // shared B fragment: B[k+2*half+{0,1}, col0+l16] (row-coalesced across lanes)
    v2f b;
    b.x = B[(size_t)(k + 2 * half + 0) * N + col0 + l16];
    b.y = B[(size_t)(k + 2 * half + 1) * N + col0 + l16];
    // four A fragments, contiguous float2 each
    v2f a0 = *(const v2f*)(Arow + k);
    v2f a1 = *(const v2f*)(Arow + k + rstep);
    v2f a2 = *(const v2f*)(Arow + k + 2 * rstep);
    v2f a3 = *(const v2f*)(Arow + k + 3 * rstep);
    acc0 = __builtin_amdgcn_wmma_f32_16x16x4_f32(false, a0, false, b, (short)0, acc0, false, false);
    acc1 = __builtin_amdgcn_wmma_f32_16x16x4_f32(false, a1, false, b, (short)0, acc1, false, false);
    acc2 = __builtin_amdgcn_wmma_f32_16x16x4_f32(false, a2, false, b, (short)0, acc2, false, false);
    acc3 = __builtin_amdgcn_wmma_f32_16x16x4_f32(false, a3, false, b, (short)0, acc3, false, false);
  }
  // C/D layout: VGPR v of lane (half,l16) = C[tile_row0 + v + 8*half, col0 + l16]
  float* Cp = C + (size_t)(row0 + 8 * half) * N + col0 + l16;
  const size_t cstep = (size_t)16 * N;
#pragma unroll
  for (int v = 0; v < 8; ++v) {
    Cp[(size_t)v * N]             = acc0[v];
    Cp[(size_t)v * N +     cstep] = acc1[v];
    Cp[(size_t)v * N + 2 * cstep] = acc2[v];
    Cp[(size_t)v * N + 3 * cstep] = acc3[v];
  }
}

// ---------------- q_sqrt = tril(exp(0.5 * q_log_var)) ----------------
__global__ void qsqrt_kernel(const float* __restrict__ qlv, float* __restrict__ qs, int m) {
  int idx = blockIdx.x * blockDim.x + threadIdx.x;
  if (idx >= m * m) return;
  int i = idx / m, j = idx % m;
  qs[(size_t)i * m + j] = (j <= i) ? expf(0.5f * qlv[(size_t)i * m + j]) : 0.0f;
}

// ---------------- q_cov = qs @ qs^T + jitter*I ----------------
__global__ void qcov_kernel(const float* __restrict__ qs, float* __restrict__ qcov, int m) {
  int idx = blockIdx.x * blockDim.x + threadIdx.x;
  if (idx >= m * m) return;
  int i = idx / m, j = idx % m;
  int kmax = (i < j ? i : j) + 1;          // both factors lower-triangular
  float s = 0.0f;
  for (int k = 0; k < kmax; ++k)
    s += qs[(size_t)i * m + k] * qs[(size_t)j * m + k];
  if (i == j) s += JITTER;
  qcov[(size_t)i * m + j] = s;
}

// ---------------- per-column ELBO contribution ----------------
// diag(pred_cov)[i] = outputscale - sum_u Lam[u,i]^2 + sum_u W[u,i]^2, W = q_cov @ Lam
__global__ void per_column_kernel(const float* __restrict__ Lam, const float* __restrict__ W,
                                  const float* __restrict__ qmu, const float* __restrict__ y,
                                  const float* __restrict__ os_p, const float* __restrict__ llv_p,
                                  float* __restrict__ contrib, int m, int n) {
  int i = blockIdx.x * blockDim.x + threadIdx.x;
  if (i >= n) return;
  float pm = 0.0f, ss = 0.0f, vs = 0.0f;
  for (int u = 0; u < m; ++u) {
    float l = Lam[(size_t)u * n + i];
    pm += l * qmu[u];
    ss += l * l;
    float w = W[(size_t)u * n + i];
    vs += w * w;
  }
  float sig = expf(llv_p[0]);
  float diagcov = os_p[0] - ss + vs;
  float diff = y[i] - pm;
  contrib[i] = -0.5f * (diff * diff + diagcov) / sig - 0.5f * logf(TWO_PI_F * sig);
}

// ---------------- final scalar: ell - kl ----------------
__global__ void finalize_kernel(const float* __restrict__ contrib, const float* __restrict__ qcov,
                                const float* __restrict__ Lq, const float* __restrict__ qmu,
                                float* __restrict__ out, int n, int m) {
  __shared__ float red[256];
  int tid = threadIdx.x;
  float s = 0.0f;
  for (int i = tid; i < n; i += 256) s += contrib[i];
  float tr = 0.0f, mmu = 0.0f, ld = 0.0f;
  for (int j = tid; j < m; j += 256) {
    tr  += qcov[(size_t)j * m + j];
    mmu += qmu[j] * qmu[j];
    ld  += logf(Lq[(size_t)j * m + j]);
  }
  auto block_sum = [&](float v) -> float {
    red[tid] = v; __syncthreads();
    for (int off = 128; off > 0; off >>= 1) {
      if (tid < off) red[tid] += red[tid + off];
      __syncthreads();
    }
    float r = red[0]; __syncthreads();
    return r;
  };
  float ellsum = block_sum(s);
  float trs    = block_sum(tr);
  float mms    = block_sum(mmu);
  float lds    = block_sum(ld);
  if (tid == 0) {
    float ell = ellsum / (float)n;
    float logdet_q = 2.0f * lds;
    float kl = 0.5f * (trs + mms - (float)m - logdet_q) / (float)n;
    out[0] = ell - kl;
  }
}

// ---------------- host-side launcher ----------------
extern "C" void kernel_launch(void* const* d_in, const int* in_sizes, int n_in,
                              void* d_out, int out_size, void* d_ws, size_t ws_size,
                              hipStream_t stream) {
  const float* train_x = (const float*)d_in[0];
  const float* train_y = (const float*)d_in[1];
  const float* Z       = (const float*)d_in[2];
  const float* q_mu    = (const float*)d_in[3];
  const float* q_lvar  = (const float*)d_in[4];
  const float* llv     = (const float*)d_in[5];
  const float* ls      = (const float*)d_in[6];
  const float* os      = (const float*)d_in[7];
  float* out = (float*)d_out;

  const int n = in_sizes[1];          // 8192
  const int m = in_sizes[3];          // 512
  const int d = in_sizes[0] / n;      // 8

  // workspace layout (floats)
  float* ws      = (float*)d_ws;
  float* Kuu     = ws;                                  // m*m  (becomes L in-place)
  float* Linv    = Kuu  + (size_t)m * m;                // m*m
  float* Kuf     = Linv + (size_t)m * m;                // m*n
  float* Lam     = Kuf  + (size_t)m * n;                // m*n
  float* qs      = Lam  + (size_t)m * n;                // m*m
  float* qcov    = qs   + (size_t)m * m;                // m*m
  float* Lq      = qcov + (size_t)m * m;                // m*m
  float* W       = Lq   + (size_t)m * m;                // m*n
  float* contrib = W    + (size_t)m * n;                // n

  const int T = 256;
  // 1) K_uu with jitter
  rbf_kuu_kernel<<<(m * m + T - 1) / T, T, 0, stream>>>(Z, Kuu, ls, os, m, d);
  // 2) K_uf
  {
    long long tot = (long long)m * n;
    rbf_kuf_kernel<<<(unsigned)((tot + T - 1) / T), T, 0, stream>>>(Z, train_x, Kuf, ls, os, m, n, d);
  }
  // 3) L = chol(K_uu) in place
  cholesky_kernel<<<1, T, 0, stream>>>(Kuu, m);
  // 4) Linv = L^{-1}  (one wave32 per column)
  trinv_kernel<<<m, 32, m * sizeof(float), stream>>>(Kuu, Linv, m);
  // 5) Lam = Linv @ Kuf   (m x n, WMMA f32, 64x16 per wave)
  {
    int tasks = (m >> 6) * (n >> 4);
    wmma_gemm_f32_kernel<<<(tasks + 7) / 8, 256, 0, stream>>>(Linv, Kuf, Lam, m, n, m);
  }
  // 6) q_sqrt, q_cov
  qsqrt_kernel<<<(m * m + T - 1) / T, T, 0, stream>>>(q_lvar, qs, m);
  qcov_kernel<<<(m * m + T - 1) / T, T, 0, stream>>>(qs, qcov, m);
  // 7) Lq = chol(q_cov)
  hipMemcpyAsync(Lq, qcov, (size_t)m * m * sizeof(float), hipMemcpyDeviceToDevice, stream);
  cholesky_kernel<<<1, T, 0, stream>>>(Lq, m);
  // 8) W = q_cov @ Lam   (m x n, WMMA f32, 64x16 per wave)
  {
    int tasks = (m >> 6) * (n >> 4);
    wmma_gemm_f32_kernel<<<(tasks + 7) / 8, 256, 0, stream>>>(qcov, Lam, W, m, n, m);
  }
  // 9) per-column ELBO terms, then scalar combine
  per_column_kernel<<<(n + T - 1) / T, T, 0, stream>>>(Lam, W, q_mu, train_y, os, llv,
                                                       contrib, m, n);
  finalize_kernel<<<1, 256, 0, stream>>>(contrib, qcov, Lq, q_mu, out, n, m);
}